// SpeculativeQSGPipeline_82497731821693
// MI455X (gfx1250) — compile-verified
//
#include <hip/hip_runtime.h>
#include <hip/hip_bf16.h>
#include <stdint.h>

// Problem constants (from reference): B=8, S=512, V=50257, K=4 drafts, L=8 len.
#define B_      8
#define S_      512
#define V_      50257
#define K_      4
#define L_      8
#define KL      32            // K_*L_ = one wave per sample
#define THRESH_ 0.8f
#define TILE    4096          // floats per LDS tile (16 KB)
#define NTILES  ((V_ + TILE - 1) / TILE)   // 13
#define THREADS 1024          // 32 waves (wave32)

typedef __attribute__((ext_vector_type(4))) unsigned int v4u;
typedef __attribute__((ext_vector_type(4))) int          v4i;
typedef __attribute__((ext_vector_type(8))) int          v8i;

#if __has_builtin(__builtin_amdgcn_s_wait_tensorcnt)
#define WAIT_TENSORCNT(n) __builtin_amdgcn_s_wait_tensorcnt((short)(n))
#else
#define WAIT_TENSORCNT(n) asm volatile("s_wait_tensorcnt %0" :: "i"(n) : "memory")
#endif

#if __has_builtin(__builtin_amdgcn_s_wait_asynccnt)
#define WAIT_ASYNCCNT(n) __builtin_amdgcn_s_wait_asynccnt(n)
#else
#define WAIT_ASYNCCNT(n) asm volatile("s_wait_asynccnt %0" :: "i"(n) : "memory")
#endif

#if __has_builtin(__builtin_amdgcn_tensor_load_to_lds)
#define HAVE_TDM 1
#else
#define HAVE_TDM 0
#endif

#if HAVE_TDM
// Issue one TDM transfer: nElems contiguous f32 from gsrc into LDS byte
// offset ldsOff. Descriptor per CDNA5 ISA.
__device__ __forceinline__ void tdm_issue(const float* gsrc, unsigned ldsOff, int nElems) {
  unsigned long long ga = (unsigned long long)(uintptr_t)gsrc;
  v4u g0;
  g0.x = 1u;                                            // count=1 (valid user D#)
  g0.y = ldsOff;                                        // lds_addr [63:32]
  g0.z = (unsigned)(ga & 0xffffffffu);                  // global_addr lo
  g0.w = (unsigned)((ga >> 32) & 0x01ffffffu)           // global_addr hi (57b total)
       | (2u << 30);                                    // type = 2 ("image")
  v8i g1;
  g1[0] = (int)(2u << 16);                              // data_size=2 (4B), mask=0
  g1[1] = (int)(((unsigned)V_ & 0xffffu) << 16);        // tensor_dim0[15:0] @ bits63:48
  g1[2] = (int)(((unsigned)V_ >> 16) | (1u << 16));     // tensor_dim0[31:16] | tensor_dim1=1
  g1[3] = (int)(((unsigned)nElems) << 16);              // tensor_dim1 hi=0 | tile_dim0
  g1[4] = 1;                                            // tile_dim1=1, tile_dim2=0
  g1[5] = (int)V_;                                      // tensor_dim0_stride[31:0]
  g1[6] = 0;                                            // stride0 hi | stride1 lo
  g1[7] = 0;
  v4i z4 = {0, 0, 0, 0};                                // dims 3/4 unused
#if defined(__clang_major__) && (__clang_major__ >= 23)
  v8i z8 = {0, 0, 0, 0, 0, 0, 0, 0};
  __builtin_amdgcn_tensor_load_to_lds(g0, g1, z4, z4, z8, 0);
#else
  __builtin_amdgcn_tensor_load_to_lds(g0, g1, z4, z4, 0);
#endif
}
#else
// Fallback: per-lane async global->LDS copies (ASYNCcnt path).
__device__ __forceinline__ void async_ld_f32(const float* gsrc, float* ldst) {
  unsigned lds_off = (unsigned)(uintptr_t)ldst;   // addr[31:0] == LDS byte offset
  asm volatile("global_load_async_to_lds_b32 %0, %1, off"
               :: "v"(lds_off), "v"(gsrc)
               : "memory");
}
#endif

__device__ __forceinline__ unsigned hash_mix(unsigned h) {
  h ^= h >> 16; h *= 0x7feb352du;
  h ^= h >> 15; h *= 0x846ca68bu;
  h ^= h >> 16;
  return h;
}

__global__ __launch_bounds__(THREADS)
void spec_qsg_kernel(const float* __restrict__ logits, float* __restrict__ out) {
  const int b    = blockIdx.x;
  const int tid  = threadIdx.x;
  const int w    = tid >> 5;    // wave id == sample id (k*L + l)
  const int lane = tid & 31;

  __shared__ float tileBuf[2][TILE];   // 32 KB double buffer
  __shared__ float sM, sZ;             // softmax max / denom
  __shared__ int   sTok[KL];
  __shared__ float sZl[KL];            // logit value at sampled token

  const float* rowBase = logits + ((size_t)b * S_ + (S_ - 1)) * (size_t)V_;

#if HAVE_TDM
  const bool issuer = (w == 0);        // one wave drives the TDM
  const unsigned ldsOff0 =
      (unsigned)__builtin_amdgcn_readfirstlane((int)(uintptr_t)(void*)&tileBuf[0][0]);
  const unsigned ldsOff1 =
      (unsigned)__builtin_amdgcn_readfirstlane((int)(uintptr_t)(void*)&tileBuf[1][0]);
  auto issue = [&](int t, int bufSel) {
    const int rem = (V_ - t * TILE) < TILE ? (V_ - t * TILE) : TILE;
    tdm_issue(rowBase + (size_t)t * TILE, bufSel ? ldsOff1 : ldsOff0, rem);
  };
#else
  auto issue = [&](int t, int bufSel) {
    #pragma unroll
    for (int j = 0; j < 4; ++j) {
      int idx = t * TILE + j * THREADS + tid;
      int c   = idx < V_ ? idx : (V_ - 1);   // clamp in-row, masked later
      async_ld_f32(rowBase + c, &tileBuf[bufSel][j * THREADS + tid]);
    }
  };
#endif

  // Per-wave Gumbel-argmax tracker; wave 0 also does online softmax.
  float bestScore = -3.4e38f;
  int   bestV     = 0;
  float bestZ     = -3.4e38f;
  float m = -3.4e38f, ssum = 0.f;

#if HAVE_TDM
  if (issuer) issue(0, 0);
#else
  issue(0, 0);
#endif

  for (int t = 0; t < NTILES; ++t) {
    const int cur = t & 1;
#if HAVE_TDM
    if (issuer) {
      if (t + 1 < NTILES) {
        issue(t + 1, cur ^ 1);    // overlap next tile's DMA with this tile's compute
        WAIT_TENSORCNT(1);        // tile t retired (TDM ops in-order per wave)
      } else {
        WAIT_TENSORCNT(0);
      }
    }
#else
    if (t + 1 < NTILES) {
      issue(t + 1, cur ^ 1);
      WAIT_ASYNCCNT(4);
    } else {
      WAIT_ASYNCCNT(0);
    }
#endif
    __syncthreads();              // tile t fully resident in LDS for all waves

    const float* buf   = tileBuf[cur];
    const int    tbase = t * TILE;
    #pragma unroll 1
    for (int c = 0; c < TILE / 128; ++c) {   // 32 chunks, ds_load_b128 each
      const int e = c * 128 + lane * 4;
      float4 x4 = *(const float4*)(buf + e);
      #pragma unroll
      for (int j = 0; j < 4; ++j) {
        const int v = tbase + e + j;
        const float x = (j == 0) ? x4.x : (j == 1) ? x4.y : (j == 2) ? x4.z : x4.w;
        if (v < V_) {
          // deterministic per-(b, sample, v) uniform -> Gumbel noise
          unsigned h = hash_mix((unsigned)v * 0x9e3779b1u
                              + (unsigned)w * 0x85ebca77u
                              + (unsigned)b * 0xc2b2ae3du);
          float u  = (float)(h >> 8) * 0x1p-24f + 0x1p-25f;  // (0,1)
          float g  = -__logf(-__logf(u));
          float sc = x + g;                                  // TEMPERATURE == 1
          if (sc > bestScore) { bestScore = sc; bestV = v; bestZ = x; }
          if (w == 0) {                                      // online softmax
            float mo = m;
            m    = fmaxf(m, x);
            ssum = ssum * __expf(mo - m) + __expf(x - m);
          }
        }
      }
    }
    __syncthreads();              // tile-t buffer free for tile t+2
  }

  // Wave-level argmax reduction (wave32: 5 shuffle steps).
  #pragma unroll
  for (int off = 16; off; off >>= 1) {
    float oS = __shfl_xor(bestScore, off);
    int   oV = __shfl_xor(bestV, off);
    float oZ = __shfl_xor(bestZ, off);
    if (oS > bestScore || (oS == bestScore && oV < bestV)) {
      bestScore = oS; bestV = oV; bestZ = oZ;
    }
  }
  if (w == 0) {                   // softmax reduction across wave 0's lanes
    #pragma unroll
    for (int off = 16; off; off >>= 1) {
      float om = __shfl_xor(m, off);
      float os = __shfl_xor(ssum, off);
      float nm = fmaxf(m, om);
      ssum = ssum * __expf(m - nm) + os * __expf(om - nm);
      m = nm;
    }
    if (lane == 0) { sM = m; sZ = ssum; }
  }
  if (lane == 0) { sTok[w] = bestV; sZl[w] = bestZ; }
  __syncthreads();

  // Wave 0 finalizes all outputs for this batch row.
  if (tid < KL) {
    const int   tok  = sTok[tid];
    const float prob = __expf(sZl[tid] - sM) / sZ;
    const float acc  = (prob >= THRESH_) ? 1.f : 0.f;

    out[              b * KL + tid] = (float)tok;   // draft_tokens   [0,256)
    out[    B_ * KL + b * KL + tid] = prob;         // draft_probs    [256,512)
    out[2 * B_ * KL + b * KL + tid] = acc;          // accepted_mask  [512,768)

    float ps = prob, as = acc;                      // reduce over L within draft
    #pragma unroll
    for (int off = 4; off; off >>= 1) {
      ps += __shfl_xor(ps, off);
      as += __shfl_xor(as, off);
    }
    const float meanp = ps * (1.f / L_);
    const float ratio = as * (1.f / L_);
    if ((tid & 7) == 0)
      out[3 * B_ * KL + b * K_ + (tid >> 3)] = ratio;   // acceptance_ratio [768,800)

    const float m0 = __shfl(meanp, 0),  m1 = __shfl(meanp, 8);
    const float m2 = __shfl(meanp, 16), m3 = __shfl(meanp, 24);
    if (tid == 0) {
      int best = 0; float bm = m0;
      if (m1 > bm) { bm = m1; best = 1; }
      if (m2 > bm) { bm = m2; best = 2; }
      if (m3 > bm) { bm = m3; best = 3; }
      out[3 * B_ * KL + B_ * K_ + b] = (float)best;     // best_draft_idx [800,808)
    }
  }
}

extern "C" void kernel_launch(void* const* d_in, const int* in_sizes, int n_in,
                              void* d_out, int out_size, void* d_ws, size_t ws_size,
                              hipStream_t stream) {
  (void)in_sizes; (void)n_in; (void)d_ws; (void)ws_size; (void)out_size;
  // d_in[0] = hidden_states (dead in reference), d_in[1] = logits,
  // d_in[2] = verifier_logits (dead in reference).
  const float* logits = (const float*)d_in[1];
  float* out = (float*)d_out;
  spec_qsg_kernel<<<dim3(B_), dim3(THREADS), 0, stream>>>(logits, out);
}